// DecoderUnDirected_65532611002918
// MI455X (gfx1250) — compile-verified
//
#include <hip/hip_runtime.h>
#include <cmath>
#include <stdint.h>

// ---------------- problem constants (from reference) ----------------
constexpr int B    = 128;
constexpr int NMAX = 64;
constexpr int PE   = 64;
constexpr int E    = 5;
constexpr int DIM  = 64;
constexpr int NF   = 16;
constexpr int TOT  = 4096;
constexpr int EP   = E * PE;        // 320
constexpr int KQW  = 2 * EP;        // 640

// output layout (floats): e_hat (B,N,N,E) | x_hat (TOT,NF) | edge_mask (B,N,N) | mask (B,N)
constexpr size_t EHAT_N = (size_t)B * NMAX * NMAX * E;   // 2,621,440
constexpr size_t XHAT_N = (size_t)TOT * NF;              //    65,536
constexpr size_t EMSK_N = (size_t)B * NMAX * NMAX;       //   524,288

typedef __attribute__((ext_vector_type(2))) float    v2f;
typedef __attribute__((ext_vector_type(8))) float    v8f;
typedef __attribute__((ext_vector_type(4))) uint32_t v4u;
typedef __attribute__((ext_vector_type(8))) uint32_t v8u;

// ---------------------------------------------------------------
// Kernel 1: pos[i] = i - start[batch[i]]  (batch sorted, one block)
// ---------------------------------------------------------------
__global__ void pos_kernel(const int* __restrict__ batch, int* __restrict__ pos) {
  __shared__ int cnts[B];
  __shared__ int st[B];
  for (int i = threadIdx.x; i < B; i += blockDim.x) cnts[i] = 0;
  __syncthreads();
  for (int i = threadIdx.x; i < TOT; i += blockDim.x) atomicAdd(&cnts[batch[i]], 1);
  __syncthreads();
  if (threadIdx.x == 0) {
    int acc = 0;
    for (int j = 0; j < B; ++j) { st[j] = acc; acc += cnts[j]; }
  }
  __syncthreads();
  for (int i = threadIdx.x; i < TOT; i += blockDim.x) pos[i] = i - st[batch[i]];
}

// ---------------------------------------------------------------
// Kernel 2: fp32 WMMA GEMM  out[M,N] = A[M,64] @ W[64,N] + bias
// Block = 256 threads = 8 waves; blockIdx.y = column tile (16 cols).
// The 64x16 W tile is staged to LDS via CDNA5 async global->LDS DMA
// (global_load_async_to_lds_b128, ASYNCcnt), then each wave computes
// one 16x16 tile with 16 x V_WMMA_F32_16X16X4_F32, B-matrix from LDS.
// Grid sized exactly -> EXEC all-ones at every WMMA (ISA requirement).
// ---------------------------------------------------------------
__global__ void gemm_wmma(const float* __restrict__ A, const float* __restrict__ W,
                          const float* __restrict__ bias, float* __restrict__ out,
                          int N /* cols, multiple of 16 */) {
  __shared__ float sW[64 * 16];               // W tile, row-major 64x16
  const int nt = blockIdx.y;
  const int t  = threadIdx.x;                 // 0..255

  // ---- async stage: 256 threads x 16B = 64 rows x 16 cols of W ----
  {
    const int r  = t >> 2;                    // W row 0..63
    const int c4 = (t & 3) << 2;              // col chunk 0,4,8,12
    const uint32_t dst  = (uint32_t)(uintptr_t)(sW + r * 16 + c4);
    const uint64_t base = (uint64_t)(uintptr_t)W;
    const uint32_t voff = (uint32_t)(((size_t)r * N + nt * 16 + c4) * sizeof(float));
    asm volatile("global_load_async_to_lds_b128 %0, %1, %2"
                 :: "v"(dst), "v"(voff), "s"(base) : "memory");
    asm volatile("s_wait_asynccnt 0x0" ::: "memory");
  }
  __syncthreads();

  const int wave = t >> 5;                    // 0..7
  const int lane = t & 31;
  const int mt   = blockIdx.x * 8 + wave;     // row tile
  const int half = lane >> 4;                 // lane group 0/1
  const int li   = lane & 15;

  const float* __restrict__ Arow = A + ((size_t)mt * 16 + li) * 64;

  v8f acc = {};
  #pragma unroll
  for (int ks = 0; ks < 16; ++ks) {
    const int kk = ks * 4 + half * 2;         // K base for this lane group
    v2f a = *(const v2f*)(Arow + kk);         // global_load_b64
    v2f b; b[0] = sW[kk * 16 + li]; b[1] = sW[(kk + 1) * 16 + li];   // ds loads
    acc = __builtin_amdgcn_wmma_f32_16x16x4_f32(false, a, false, b, (short)0, acc,
                                                false, false);
  }
  const int col = nt * 16 + li;
  const float bv = bias[col];
  #pragma unroll
  for (int vg = 0; vg < 8; ++vg) {
    const int row = mt * 16 + vg + half * 8;  // C/D layout: lanes 16-31 hold M+8
    out[(size_t)row * N + col] = acc[vg] + bv;
  }
}

// ---------------------------------------------------------------
// Kernel 3: scatter kq rows into dense per-batch K/Q + mask
// ---------------------------------------------------------------
__global__ void scatter_kernel(const float* __restrict__ kq, const int* __restrict__ batch,
                               const int* __restrict__ pos, float* __restrict__ Kd,
                               float* __restrict__ Qd, int* __restrict__ maskD) {
  const int i = blockIdx.x;             // row 0..TOT-1
  const int p = pos[i];
  if (p >= NMAX) return;                // mode='drop'
  const int b = batch[i];
  const size_t dst = ((size_t)b * NMAX + p) * EP;
  const float* __restrict__ src = kq + (size_t)i * KQW;
  for (int j = threadIdx.x; j < EP; j += blockDim.x) {
    Kd[dst + j] = src[j];
    Qd[dst + j] = src[EP + j];
  }
  if (threadIdx.x == 0) maskD[b * NMAX + p] = 1;
}

// ---------------------------------------------------------------
// Kernel 4: batched Z0[b,e] = (Qh @ Kh^T) * scale via WMMA f32.
// One block (16 waves) per (b,e). Both 64x64 operand blocks are pulled
// into LDS by the Tensor Data Mover (tensor_load_to_lds, TENSORcnt),
// descriptors per CDNA5 ISA:
//   group0: count=1 | lds_addr | global_addr[56:0] | type=2
//   group1: data_size=4B, tensor_dim0=320, tensor_dim1=64,
//           tile_dim0=64, tile_dim1=64, tensor_dim0_stride=320
// Tile lands contiguously in LDS (64x64 f32). B-matrix of Q.K^T is K^T,
// i.e. B[k][m] = Kd[m][k] -> row-major LDS reads for both operands.
// ---------------------------------------------------------------
__global__ void z_wmma(const float* __restrict__ Qd, const float* __restrict__ Kd,
                       float* __restrict__ Z0) {
  __shared__ float sQ[NMAX * PE];       // 16 KB
  __shared__ float sK[NMAX * PE];       // 16 KB
  const int be = blockIdx.x;            // 0 .. B*E-1
  const int b = be / E, e = be - b * E;

  if (threadIdx.x < 32) {               // wave 0 drives the TDM (EXEC ignored by TDM)
    const uint64_t gaQ = (uint64_t)(uintptr_t)(Qd + ((size_t)b * NMAX) * EP + e * PE);
    const uint64_t gaK = (uint64_t)(uintptr_t)(Kd + ((size_t)b * NMAX) * EP + e * PE);
    v8u g1;
    g1[0] = (2u << 16);                         // data_size = 4 bytes
    g1[1] = ((uint32_t)EP) << 16;               // tensor_dim0[15:0] = 320
    g1[2] = ((uint32_t)NMAX) << 16;             // dim0 hi = 0 | tensor_dim1[15:0] = 64
    g1[3] = ((uint32_t)PE) << 16;               // dim1 hi = 0 | tile_dim0 = 64
    g1[4] = (uint32_t)NMAX;                     // tile_dim1 = 64, tile_dim2 = 0
    g1[5] = (uint32_t)EP;                       // tensor_dim0_stride = 320
    g1[6] = 0u;
    g1[7] = 0u;
    v4u g0q, g0k;
    g0q[0] = 1u;                                // count=1, user mode
    g0q[1] = (uint32_t)(uintptr_t)sQ;           // lds_addr
    g0q[2] = (uint32_t)gaQ;
    g0q[3] = (uint32_t)((gaQ >> 32) & 0x01FFFFFFu) | 0x80000000u;   // type=2
    g0k[0] = 1u;
    g0k[1] = (uint32_t)(uintptr_t)sK;
    g0k[2] = (uint32_t)gaK;
    g0k[3] = (uint32_t)((gaK >> 32) & 0x01FFFFFFu) | 0x80000000u;
    asm volatile("tensor_load_to_lds %0, %1" :: "s"(g0q), "s"(g1) : "memory");
    asm volatile("tensor_load_to_lds %0, %1" :: "s"(g0k), "s"(g1) : "memory");
    __builtin_amdgcn_s_wait_tensorcnt(0);
  }
  __syncthreads();

  const int wave = threadIdx.x >> 5;    // 0..15
  const int lane = threadIdx.x & 31;
  const int tr = wave >> 2, tc = wave & 3;
  const int half = lane >> 4, li = lane & 15;

  const float* Qrow = sQ + (tr * 16 + li) * PE;
  const float* Krow = sK + (tc * 16 + li) * PE;

  v8f acc = {};
  #pragma unroll
  for (int ks = 0; ks < 16; ++ks) {
    const int kk = ks * 4 + half * 2;
    v2f a  = *(const v2f*)(Qrow + kk);          // ds_load_b64
    v2f bb = *(const v2f*)(Krow + kk);          // ds_load_b64
    acc = __builtin_amdgcn_wmma_f32_16x16x4_f32(false, a, false, bb, (short)0, acc,
                                                false, false);
  }
  const float scale = 0.05590169943749474f;     // 1/sqrt(320)
  const int m = tc * 16 + li;
  #pragma unroll
  for (int vg = 0; vg < 8; ++vg) {
    const int n = tr * 16 + vg + half * 8;
    Z0[(((size_t)be * NMAX) + n) * NMAX + m] = acc[vg] * scale;
  }
}

// ---------------------------------------------------------------
// Kernel 5: masked row sums of symmetrized z; neighbor counts; mask out
// thread per (b,e,n)
// ---------------------------------------------------------------
__global__ void rowsum_kernel(const float* __restrict__ Z0, const int* __restrict__ maskD,
                              float* __restrict__ Sz, float* __restrict__ cntf,
                              float* __restrict__ out_mask) {
  const int t = blockIdx.x * blockDim.x + threadIdx.x;   // 0 .. B*E*64-1
  const int n = t & 63;
  const int be = t >> 6;
  const int e = be % E;
  const int b = be / E;
  const bool mn = maskD[b * NMAX + n] != 0;
  const float* __restrict__ zr = Z0 + ((size_t)be * NMAX + n) * NMAX;  // z[b,e,n,:]
  const float* __restrict__ zc = Z0 + (size_t)be * NMAX * NMAX + n;    // z[b,e,:,n]
  float s = 0.f; int c = 0;
  for (int m = 0; m < NMAX; ++m) {
    const bool edge = mn && (maskD[b * NMAX + m] != 0) && (m != n);
    const float zs = 0.5f * (zr[m] + zc[(size_t)m * NMAX]);
    if (edge) { s += zs; ++c; }
  }
  Sz[t] = s;
  if (e == 0) {
    cntf[b * NMAX + n] = (float)c;
    out_mask[b * NMAX + n] = mn ? 1.f : 0.f;
  }
}

// ---------------------------------------------------------------
// Kernel 6: u[k] = wctx[k,:]·phi_w ; v[k] = wctx[k,:]·phi_b  (rank-2 ctx)
// ---------------------------------------------------------------
__global__ void uv_kernel(const float* __restrict__ wctx, const float* __restrict__ phi_w,
                          const float* __restrict__ phi_b, float* __restrict__ u,
                          float* __restrict__ v) {
  const int k = threadIdx.x;            // 0..63
  float su = 0.f, sv = 0.f;
  for (int d = 0; d < DIM; ++d) {
    const float w = wctx[k * DIM + d];
    su += w * phi_w[d];
    sv += w * phi_b[d];
  }
  u[k] = su; v[k] = sv;
}

// ---------------------------------------------------------------
// Kernel 7: dominant element-wise stage (VALU-bound: 167.8M erf evals).
// thread per (b,n,m); inner loops over e (5) and d (64) with exact gelu.
// ctx[b,e,n,d] = a1*u[d] + a2*v[d]   (rank-2 simplification)
// ---------------------------------------------------------------
__global__ void edge_kernel(const float* __restrict__ Z0, const int* __restrict__ maskD,
                            const float* __restrict__ Sz, const float* __restrict__ cntf,
                            const float* __restrict__ u, const float* __restrict__ v,
                            const float* __restrict__ wself_w, const float* __restrict__ wself_b,
                            const float* __restrict__ out_w, const float* __restrict__ out_b,
                            float* __restrict__ e_hat, float* __restrict__ edge_mask_out) {
  __shared__ float sW[DIM], sB[DIM], sO[DIM], sU[DIM], sV[DIM];
  if (threadIdx.x < DIM) {
    const int d = threadIdx.x;
    sW[d] = wself_w[d]; sB[d] = wself_b[d]; sO[d] = out_w[d];
    sU[d] = u[d];       sV[d] = v[d];
  }
  __syncthreads();

  const int t = blockIdx.x * blockDim.x + threadIdx.x;   // 0 .. B*64*64-1
  const int m = t & 63;
  const int bn = t >> 6;
  const int n = bn & 63;
  const int b = bn >> 6;

  const bool edge = (maskD[b * NMAX + n] != 0) && (maskD[b * NMAX + m] != 0) && (m != n);
  edge_mask_out[t] = edge ? 1.f : 0.f;

  const float cv = cntf[bn];
  const float denom = fmaxf(cv, 1.f);
  const float a2 = cv / denom;
  const float ob = out_b[0];

  #pragma unroll
  for (int e = 0; e < E; ++e) {
    const size_t base = (size_t)(b * E + e) * NMAX;
    const float* znm = Z0 + (base + n) * NMAX + m;
    const float* zmn = Z0 + (base + m) * NMAX + n;
    __builtin_prefetch(zmn, 0, 1);                        // global_prefetch_b8
    const float zs = 0.5f * (*znm + *zmn);
    const float a1 = Sz[(b * E + e) * NMAX + n] / denom;
    float acc = 0.f;
    #pragma unroll 8
    for (int d = 0; d < DIM; ++d) {
      const float ctx = a1 * sU[d] + a2 * sV[d];
      const float pre = zs * sW[d] + sB[d] + ctx;
      const float g = 0.5f * pre * (1.f + erff(pre * 0.7071067811865476f));
      acc += g * sO[d];
    }
    e_hat[(size_t)t * E + e] = acc + ob;                  // (b,n,m,e)
  }
}

// ---------------------------------------------------------------
extern "C" void kernel_launch(void* const* d_in, const int* in_sizes, int n_in,
                              void* d_out, int out_size, void* d_ws, size_t ws_size,
                              hipStream_t stream) {
  (void)in_sizes; (void)n_in; (void)out_size; (void)ws_size;

  const float* x      = (const float*)d_in[0];
  const int*   batch  = (const int*)  d_in[1];
  const float* W_node = (const float*)d_in[2];
  const float* b_node = (const float*)d_in[3];
  const float* W_attn = (const float*)d_in[4];
  const float* b_attn = (const float*)d_in[5];
  const float* phi_w  = (const float*)d_in[6];
  const float* phi_b  = (const float*)d_in[7];
  const float* wselfw = (const float*)d_in[8];
  const float* wselfb = (const float*)d_in[9];
  const float* wctx   = (const float*)d_in[10];
  const float* out_w  = (const float*)d_in[11];
  const float* out_b  = (const float*)d_in[12];

  float* out   = (float*)d_out;
  float* ehat  = out;                       // (B,N,N,E)
  float* xhat  = out + EHAT_N;              // (TOT,NF)
  float* emask = xhat + XHAT_N;             // (B,N,N)
  float* maskO = emask + EMSK_N;            // (B,N)

  // workspace layout
  float* ws   = (float*)d_ws;
  float* kq   = ws;                         // TOT*KQW       = 2,621,440
  float* Kd   = kq + (size_t)TOT * KQW;     // B*NMAX*EP     = 2,621,440
  float* Qd   = Kd + (size_t)B * NMAX * EP;
  float* Z0   = Qd + (size_t)B * NMAX * EP; // B*E*N*N       = 2,621,440
  float* Sz   = Z0 + (size_t)B * E * NMAX * NMAX;  // 40,960
  float* cntf = Sz + (size_t)B * E * NMAX;         //  8,192
  float* uA   = cntf + (size_t)B * NMAX;           //     64
  float* vA   = uA + DIM;                          //     64
  int*   maskD = (int*)(vA + DIM);                 //  8,192 ints
  int*   posA  = maskD + B * NMAX;                 //  4,096 ints

  // dense K/Q and mask must be zero each call (harness does not re-poison)
  hipMemsetAsync(Kd, 0, (size_t)2 * B * NMAX * EP * sizeof(float), stream);
  hipMemsetAsync(maskD, 0, (size_t)B * NMAX * sizeof(int), stream);

  pos_kernel<<<1, 1024, 0, stream>>>(batch, posA);

  // kq = x @ W_attn + b_attn : 32 row-blocks x 40 col tiles
  gemm_wmma<<<dim3(TOT / 16 / 8, KQW / 16), 256, 0, stream>>>(x, W_attn, b_attn, kq, KQW);
  // x_hat = x @ W_node + b_node : 32 row-blocks x 1 col tile
  gemm_wmma<<<dim3(TOT / 16 / 8, NF / 16), 256, 0, stream>>>(x, W_node, b_node, xhat, NF);

  scatter_kernel<<<TOT, 256, 0, stream>>>(kq, batch, posA, Kd, Qd, maskD);

  // Z0: one block (16 waves) per (b,e), operands staged by TDM
  z_wmma<<<B * E, 512, 0, stream>>>(Qd, Kd, Z0);

  rowsum_kernel<<<(B * E * NMAX) / 256, 256, 0, stream>>>(Z0, maskD, Sz, cntf, maskO);

  uv_kernel<<<1, DIM, 0, stream>>>(wctx, phi_w, phi_b, uA, vA);

  edge_kernel<<<(B * NMAX * NMAX) / 256, 256, 0, stream>>>(
      Z0, maskD, Sz, cntf, uA, vA, wselfw, wselfb, out_w, out_b, ehat, emask);
}